// FinePoiModel_60705067762029
// MI455X (gfx1250) — compile-verified
//
#include <hip/hip_runtime.h>
#include <math.h>

// ---------------------------------------------------------------------------
// GATv2 x3 layers for MI455X (gfx1250, wave32, WMMA).
// N=50000, D=128, H=4, C=32, L=3, E=800000 (+N self loops).
// ---------------------------------------------------------------------------

#define NN 50000
#define DD 128
#define HH 4
#define CC 32
#define LL 3
#define NEG_SLOPE 0.2f
#define LN_EPS 1e-5f

typedef __attribute__((ext_vector_type(2))) float v2f;
typedef __attribute__((ext_vector_type(8))) float v8f;

// ---- order-preserving float <-> uint map for atomic segment-max ------------
__device__ __forceinline__ unsigned f2ord(float f) {
  unsigned u = __float_as_uint(f);
  return (u & 0x80000000u) ? ~u : (u | 0x80000000u);
}
__device__ __forceinline__ float ord2f(unsigned u) {
  u = (u & 0x80000000u) ? (u & 0x7FFFFFFFu) : ~u;
  return __uint_as_float(u);
}

// ---------------------------------------------------------------------------
// 0) per-layer init: zero accumulator, denom, emax
// ---------------------------------------------------------------------------
__global__ void init_layer(float* __restrict__ acc, float* __restrict__ denom,
                           unsigned* __restrict__ emax, int nd, int nh) {
  int i = blockIdx.x * blockDim.x + threadIdx.x;
  if (i < nd) acc[i] = 0.0f;
  if (i < nh) { denom[i] = 0.0f; emax[i] = 0u; }  // 0 == ord(-NaN), identity for max
}

// ---------------------------------------------------------------------------
// 1) dual GEMM via V_WMMA_F32_16X16X4_F32:
//    xl = x @ Wl + bl ; xr = x @ Wr + br
//    block = 16 rows x 128 cols, 8 waves; wave w -> cols [16w,16w+16)
// ---------------------------------------------------------------------------
__global__ __launch_bounds__(256) void gemm_dual_wmma(
    const float* __restrict__ x,
    const float* __restrict__ Wl, const float* __restrict__ bl,
    const float* __restrict__ Wr, const float* __restrict__ br,
    float* __restrict__ xl, float* __restrict__ xr) {
  __shared__ float sx[16 * DD];  // 16-row tile of x (8 KB)

  const int tid  = threadIdx.x;
  const int wave = tid >> 5;
  const int lane = tid & 31;
  const int rowBase = blockIdx.x * 16;  // N = 50000 = 3125 * 16, exact

  // cooperative LDS fill: 2048 floats = 512 float4, 256 threads x 2
  {
    const float4* gsrc = (const float4*)(x + (size_t)rowBase * DD);
    float4* sdst = (float4*)sx;
    sdst[tid]       = gsrc[tid];
    sdst[tid + 256] = gsrc[tid + 256];
  }
  __syncthreads();

  const int colBase = wave * 16;
  const int m  = lane & 15;         // A: M index
  const int kk = (lane >> 4) * 2;   // A/B: lanes 16-31 hold K+2,K+3
  const int n  = lane & 15;         // B/D: N index

  v8f cl = {};
  v8f cr = {};

  for (int k0 = 0; k0 < DD; k0 += 4) {
    v2f a;
    a.x = sx[m * DD + k0 + kk];
    a.y = sx[m * DD + k0 + kk + 1];
    v2f bfl;
    bfl.x = Wl[(size_t)(k0 + kk) * DD + colBase + n];
    bfl.y = Wl[(size_t)(k0 + kk + 1) * DD + colBase + n];
    v2f bfr;
    bfr.x = Wr[(size_t)(k0 + kk) * DD + colBase + n];
    bfr.y = Wr[(size_t)(k0 + kk + 1) * DD + colBase + n];

    cl = __builtin_amdgcn_wmma_f32_16x16x4_f32(
        false, a, false, bfl, (short)0, cl, false, false);
    cr = __builtin_amdgcn_wmma_f32_16x16x4_f32(
        false, a, false, bfr, (short)0, cr, false, false);
  }

  // D layout: VGPR i -> M=i (lanes 0-15), M=i+8 (lanes 16-31)
  const int rOff = (lane >> 4) * 8;
  const int col  = colBase + n;
  const float bL = bl[col];
  const float bR = br[col];
#pragma unroll
  for (int i = 0; i < 8; ++i) {
    const size_t row = (size_t)(rowBase + i + rOff);
    xl[row * DD + col] = cl[i] + bL;
    xr[row * DD + col] = cr[i] + bR;
  }
}

// ---------------------------------------------------------------------------
// 2) edge logits: e = att_h . leaky_relu(xl[src,h,:] + xr[dst,h,:])
//    one thread per (edge, head); atomic segment-max into emax[dst,h]
// ---------------------------------------------------------------------------
__global__ void edge_logits(const int* __restrict__ ei,
                            const float* __restrict__ xl,
                            const float* __restrict__ xr,
                            const float* __restrict__ att,
                            float* __restrict__ elog,
                            unsigned* __restrict__ emax, int E, int Np) {
  const int t = blockIdx.x * blockDim.x + threadIdx.x;
  const int tot = (E + Np) * HH;
  if (t >= tot) return;
  const int e = t >> 2;
  const int h = t & 3;
  const int src = (e < E) ? ei[e] : (e - E);
  const int dst = (e < E) ? ei[E + e] : (e - E);

  const float4* pl = (const float4*)(xl + (size_t)src * DD + h * CC);
  const float4* pr = (const float4*)(xr + (size_t)dst * DD + h * CC);
  const float4* pa = (const float4*)(att + h * CC);

  float s = 0.0f;
#pragma unroll
  for (int i = 0; i < CC / 4; ++i) {
    float4 a = pl[i], b = pr[i], w = pa[i];
    float m0 = a.x + b.x; m0 = (m0 > 0.0f) ? m0 : NEG_SLOPE * m0;
    float m1 = a.y + b.y; m1 = (m1 > 0.0f) ? m1 : NEG_SLOPE * m1;
    float m2 = a.z + b.z; m2 = (m2 > 0.0f) ? m2 : NEG_SLOPE * m2;
    float m3 = a.w + b.w; m3 = (m3 > 0.0f) ? m3 : NEG_SLOPE * m3;
    s = fmaf(w.x, m0, s);
    s = fmaf(w.y, m1, s);
    s = fmaf(w.z, m2, s);
    s = fmaf(w.w, m3, s);
  }
  elog[t] = s;
  atomicMax(&emax[dst * HH + h], f2ord(s));
}

// ---------------------------------------------------------------------------
// 3) exponentiate and accumulate softmax denominator
// ---------------------------------------------------------------------------
__global__ void edge_exp(const int* __restrict__ ei,
                         float* __restrict__ elog,
                         const unsigned* __restrict__ emax,
                         float* __restrict__ denom, int E, int Np) {
  const int t = blockIdx.x * blockDim.x + threadIdx.x;
  const int tot = (E + Np) * HH;
  if (t >= tot) return;
  const int e = t >> 2;
  const int h = t & 3;
  const int dst = (e < E) ? ei[E + e] : (e - E);
  const float mx = ord2f(emax[dst * HH + h]);
  const float ee = __expf(elog[t] - mx);
  elog[t] = ee;
  atomicAdd(&denom[dst * HH + h], ee);
}

// ---------------------------------------------------------------------------
// 4) scatter: acc[dst] += xl[src] * alpha ; one wave32 per edge, 4 chans/lane
// ---------------------------------------------------------------------------
__global__ __launch_bounds__(256) void edge_scatter(
    const int* __restrict__ ei, const float* __restrict__ xl,
    const float* __restrict__ elog, const float* __restrict__ denom,
    float* __restrict__ acc, int E, int Np) {
  const int wid  = (blockIdx.x * blockDim.x + threadIdx.x) >> 5;
  const int lane = threadIdx.x & 31;
  if (wid >= E + Np) return;
  const int src = (wid < E) ? ei[wid] : (wid - E);
  const int dst = (wid < E) ? ei[E + wid] : (wid - E);
#pragma unroll
  for (int h = 0; h < HH; ++h) {
    const float alpha = elog[wid * HH + h] / denom[dst * HH + h];
    const int c = h * CC + lane;
    atomicAdd(&acc[(size_t)dst * DD + c], xl[(size_t)src * DD + c] * alpha);
  }
}

// ---------------------------------------------------------------------------
// 5) residual + LayerNorm (+ optional ReLU); one wave32 per node, 4 chans/lane
// ---------------------------------------------------------------------------
__global__ __launch_bounds__(256) void node_norm(
    const float* __restrict__ xin, const float* __restrict__ acc,
    const float* __restrict__ bias, const float* __restrict__ gamma,
    const float* __restrict__ beta, float* __restrict__ xout, int Np,
    int do_relu) {
  const int wid  = (blockIdx.x * blockDim.x + threadIdx.x) >> 5;
  const int lane = threadIdx.x & 31;
  if (wid >= Np) return;

  float v[4];
  float sum = 0.0f;
#pragma unroll
  for (int j = 0; j < 4; ++j) {
    const int c = j * 32 + lane;
    const float t = xin[(size_t)wid * DD + c] + acc[(size_t)wid * DD + c] + bias[c];
    v[j] = t;
    sum += t;
  }
#pragma unroll
  for (int off = 16; off > 0; off >>= 1) sum += __shfl_xor(sum, off, 32);
  const float mu = sum * (1.0f / (float)DD);

  float sq = 0.0f;
#pragma unroll
  for (int j = 0; j < 4; ++j) { const float d = v[j] - mu; sq = fmaf(d, d, sq); }
#pragma unroll
  for (int off = 16; off > 0; off >>= 1) sq += __shfl_xor(sq, off, 32);
  const float rstd = rsqrtf(sq * (1.0f / (float)DD) + LN_EPS);

#pragma unroll
  for (int j = 0; j < 4; ++j) {
    const int c = j * 32 + lane;
    float o = (v[j] - mu) * rstd * gamma[c] + beta[c];
    if (do_relu) o = (o > 0.0f) ? o : 0.0f;
    xout[(size_t)wid * DD + c] = o;
  }
}

// ---------------------------------------------------------------------------
// host launcher
// ---------------------------------------------------------------------------
static inline int cdiv(long long a, long long b) { return (int)((a + b - 1) / b); }

extern "C" void kernel_launch(void* const* d_in, const int* in_sizes, int n_in,
                              void* d_out, int out_size, void* d_ws, size_t ws_size,
                              hipStream_t stream) {
  const int Np = NN;
  const int E  = in_sizes[1] / 2;
  const int EP = E + Np;

  const float* x0    = (const float*)d_in[0];
  const int*   ei    = (const int*)d_in[1];
  const float* Wl    = (const float*)d_in[2];
  const float* bl    = (const float*)d_in[3];
  const float* Wr    = (const float*)d_in[4];
  const float* br    = (const float*)d_in[5];
  const float* att   = (const float*)d_in[6];
  const float* bias  = (const float*)d_in[7];
  const float* gamma = (const float*)d_in[8];
  const float* beta  = (const float*)d_in[9];

  // workspace layout (floats)
  float* ws = (float*)d_ws;
  const size_t ND = (size_t)Np * DD;
  const size_t NH = (size_t)Np * HH;
  float*    bufA  = ws;                    // ping buffer for x between layers
  float*    xl    = bufA + ND;
  float*    xr    = xl + ND;
  float*    acc   = xr + ND;
  float*    elog  = acc + ND;              // EP * HH
  float*    denom = elog + (size_t)EP * HH;
  unsigned* emax  = (unsigned*)(denom + NH);

  const int TB = 256;
  const dim3 blk(TB);

  for (int i = 0; i < LL; ++i) {
    const float* xin  = (i == 0) ? x0 : bufA;
    float*       xout = (i == LL - 1) ? (float*)d_out : bufA;
    const float* Wli  = Wl + (size_t)i * DD * DD;
    const float* bli  = bl + (size_t)i * DD;
    const float* Wri  = Wr + (size_t)i * DD * DD;
    const float* bri  = br + (size_t)i * DD;
    const float* atti = att + (size_t)i * HH * CC;
    const float* bi   = bias + (size_t)i * DD;
    const float* gi   = gamma + (size_t)i * DD;
    const float* bti  = beta + (size_t)i * DD;

    init_layer<<<cdiv(ND, TB), blk, 0, stream>>>(acc, denom, emax, (int)ND, (int)NH);

    gemm_dual_wmma<<<Np / 16, blk, 0, stream>>>(xin, Wli, bli, Wri, bri, xl, xr);

    edge_logits<<<cdiv((long long)EP * HH, TB), blk, 0, stream>>>(
        ei, xl, xr, atti, elog, emax, E, Np);

    edge_exp<<<cdiv((long long)EP * HH, TB), blk, 0, stream>>>(
        ei, elog, emax, denom, E, Np);

    edge_scatter<<<cdiv((long long)EP * 32, TB), blk, 0, stream>>>(
        ei, xl, elog, denom, acc, E, Np);

    node_norm<<<cdiv((long long)Np * 32, TB), blk, 0, stream>>>(
        xin, acc, bi, gi, bti, xout, Np, (i != LL - 1) ? 1 : 0);
  }
}